// LiveNet_83923660963904
// MI455X (gfx1250) — compile-verified
//
#include <hip/hip_runtime.h>

// Problem sizes (fixed by the reference)
#define N_SRC   50000
#define N_DST   50000
#define N_EDGES 1600000
#define BATCH   64

// Scatter kernel config
#define CHUNK        1024                 // edges staged per chunk (words per array)
#define SCAT_BLOCKS  512
#define SCAT_THREADS 256
#define NWAVES       (SCAT_THREADS / 32)  // 8 waves (wave32)
#define EPW          (CHUNK / NWAVES)     // 128 edges per wave per chunk
#define NC_FULL      (N_EDGES / CHUNK)    // 1562 full chunks
#define TAIL_START   (NC_FULL * CHUNK)    // 1,599,488 (tail = 512 edges)

// ---------------------------------------------------------------------------
// CDNA5 async global->LDS copy (ASYNCcnt path). VDST = LDS byte address VGPR,
// VADDR = 32-bit per-lane byte offset, SADDR = 64-bit SGPR base (GVS mode).
// ---------------------------------------------------------------------------
__device__ __forceinline__ void async_b128(unsigned lds_addr, unsigned voff,
                                           const void* sbase) {
    asm volatile("global_load_async_to_lds_b128 %0, %1, %2"
                 :: "v"(lds_addr), "v"(voff), "s"(sbase)
                 : "memory");
}

__device__ __forceinline__ void wait_async_le3() {
    asm volatile("s_wait_asynccnt 3" ::: "memory");
}
__device__ __forceinline__ void wait_async_0() {
    asm volatile("s_wait_asynccnt 0" ::: "memory");
}

__device__ __forceinline__ void atomic_add_f32(float* p, float v) {
    // hardware global_atomic_add_f32 (no CAS loop)
    unsafeAtomicAdd(p, v);
}

// ---------------------------------------------------------------------------
// Kernel 0: zero the accumulator agg[N_DST][64] in workspace (float4 stores)
// ---------------------------------------------------------------------------
__global__ void zero_agg(float4* __restrict__ agg4, int n4) {
    int i = blockIdx.x * blockDim.x + threadIdx.x;
    int stride = gridDim.x * blockDim.x;
    float4 z = make_float4(0.f, 0.f, 0.f, 0.f);
    for (; i < n4; i += stride) agg4[i] = z;
}

// ---------------------------------------------------------------------------
// Kernel 1: transpose x[64][N_SRC] -> xT[N_SRC][64] (LDS-tiled, conflict-free)
// One block handles 32 source columns x 64 batch rows.
// ---------------------------------------------------------------------------
__global__ void transpose_x(const float* __restrict__ x, float* __restrict__ xT) {
    __shared__ float tile[32 * 65];
    const int s0 = blockIdx.x * 32;
    const int t  = threadIdx.x;
#pragma unroll
    for (int p = 0; p < 8; ++p) {
        int i  = t + p * 256;          // 2048 elements
        int bb = i >> 5;               // batch row 0..63
        int c  = i & 31;               // column within tile
        int s  = s0 + c;
        float v = (s < N_SRC) ? x[(size_t)bb * N_SRC + s] : 0.f;
        tile[c * 65 + bb] = v;         // pad-65: banks (c+bb)%64 distinct
    }
    __syncthreads();
#pragma unroll
    for (int p = 0; p < 8; ++p) {
        int j  = t + p * 256;
        int cc = j >> 6;               // column 0..31
        int bb = j & 63;               // batch 0..63
        int s  = s0 + cc;
        if (s < N_SRC) xT[((size_t)s << 6) + bb] = tile[cc * 65 + bb];
    }
}

// ---------------------------------------------------------------------------
// Kernel 2: edge scatter. Double-buffered async staging of (src,dst,k) chunks
// into LDS; one wave32 per edge: lane l covers batch {2l, 2l+1} via float2,
// 2 coalesced f32 atomics per lane into agg[dst][64].
// ---------------------------------------------------------------------------
__global__ void __launch_bounds__(SCAT_THREADS)
edge_scatter(const int*   __restrict__ edge_src,
             const int*   __restrict__ edge_dst,
             const float* __restrict__ kw,
             const float* __restrict__ xT,
             float*       __restrict__ agg) {
    __shared__ __align__(16) int   sbuf[2][CHUNK];
    __shared__ __align__(16) int   dbuf[2][CHUNK];
    __shared__ __align__(16) float kbuf[2][CHUNK];

    const int t    = threadIdx.x;
    const int wv   = t >> 5;
    const int lane = t & 31;

    // Stage chunk cc into buffer bf: each thread issues 3 x B128 (4 words each)
    auto stage = [&](int cc, int bf) {
        unsigned gb = (unsigned)(cc * CHUNK + t * 4) * 4u;   // global byte offset
        async_b128((unsigned)(size_t)&sbuf[bf][t * 4], gb, (const void*)edge_src);
        async_b128((unsigned)(size_t)&dbuf[bf][t * 4], gb, (const void*)edge_dst);
        async_b128((unsigned)(size_t)&kbuf[bf][t * 4], gb, (const void*)kw);
    };

    const int grid = gridDim.x;
    int c = blockIdx.x;
    int bufc = 0;
    if (c < NC_FULL) stage(c, 0);       // prologue

    for (; c < NC_FULL; c += grid) {
        int cn = c + grid;
        if (cn < NC_FULL) {
            stage(cn, bufc ^ 1);        // overlap next chunk's DMA with compute
            wait_async_le3();           // previous 3 (this chunk) are complete
        } else {
            wait_async_0();
        }
        __syncthreads();                // all waves' staged data visible

        const int base = wv * EPW;
#pragma unroll 4
        for (int j = 0; j < EPW; ++j) {
            int   s = sbuf[bufc][base + j];   // wave-uniform -> LDS broadcast
            int   d = dbuf[bufc][base + j];
            float w = kbuf[bufc][base + j];
            const float2 xv =
                *(const float2*)(xT + ((size_t)s << 6) + (lane << 1));
            size_t o = ((size_t)d << 6) + (size_t)(lane << 1);
            atomic_add_f32(&agg[o],     w * xv.x);
            atomic_add_f32(&agg[o + 1], w * xv.y);
        }
        __syncthreads();                // buffer reusable
        bufc ^= 1;
    }

    // Tail edges (512): plain loads, one wave per edge
    int gwave = (int)((blockIdx.x * blockDim.x + threadIdx.x) >> 5);
    int totw  = (int)((gridDim.x * blockDim.x) >> 5);
    for (int e = TAIL_START + gwave; e < N_EDGES; e += totw) {
        int   s = edge_src[e];
        int   d = edge_dst[e];
        float w = kw[e];
        const float2 xv = *(const float2*)(xT + ((size_t)s << 6) + (lane << 1));
        size_t o = ((size_t)d << 6) + (size_t)(lane << 1);
        atomic_add_f32(&agg[o],     w * xv.x);
        atomic_add_f32(&agg[o + 1], w * xv.y);
    }
}

// ---------------------------------------------------------------------------
// Kernel 3: out[b][d] = relu(agg[d][b] + bias[d])  (LDS-tiled transpose back)
// One block handles 32 destinations x 64 batch.
// ---------------------------------------------------------------------------
__global__ void finalize(const float* __restrict__ agg,
                         const float* __restrict__ bias,
                         float* __restrict__ out) {
    __shared__ float tile[32 * 65];
    const int d0 = blockIdx.x * 32;
    const int t  = threadIdx.x;
#pragma unroll
    for (int p = 0; p < 8; ++p) {
        int i  = t + p * 256;          // contiguous agg reads
        int r  = i >> 6;
        int bb = i & 63;
        int d  = d0 + r;
        float v = (d < N_DST) ? agg[((size_t)d0 << 6) + i] : 0.f;
        tile[r * 65 + bb] = v;
    }
    __syncthreads();
#pragma unroll
    for (int p = 0; p < 8; ++p) {
        int j  = t + p * 256;
        int bb = j >> 5;               // batch 0..63
        int c  = j & 31;               // dest within tile
        int d  = d0 + c;
        if (d < N_DST) {
            float v = tile[c * 65 + bb] + bias[d];
            out[(size_t)bb * N_DST + d] = v > 0.f ? v : 0.f;
        }
    }
}

// ---------------------------------------------------------------------------
extern "C" void kernel_launch(void* const* d_in, const int* in_sizes, int n_in,
                              void* d_out, int out_size, void* d_ws, size_t ws_size,
                              hipStream_t stream) {
    const float* x  = (const float*)d_in[0];   // [64 * N_SRC]
    const float* kw = (const float*)d_in[1];   // [N_EDGES]
    const float* b  = (const float*)d_in[2];   // [N_DST]
    const int*   es = (const int*)d_in[3];     // [N_EDGES]
    const int*   ed = (const int*)d_in[4];     // [N_EDGES]
    float*       out = (float*)d_out;          // [64 * N_DST]

    // Workspace layout: xT (12.8 MB) | agg (12.8 MB)
    float* xT  = (float*)d_ws;
    float* agg = xT + (size_t)N_SRC * BATCH;

    zero_agg<<<2048, 256, 0, stream>>>((float4*)agg, (N_DST * BATCH) / 4);
    transpose_x<<<(N_SRC + 31) / 32, 256, 0, stream>>>(x, xT);
    edge_scatter<<<SCAT_BLOCKS, SCAT_THREADS, 0, stream>>>(es, ed, kw, xT, agg);
    finalize<<<(N_DST + 31) / 32, 256, 0, stream>>>(agg, b, out);
}